// SFDiVeQ_78426102825290
// MI455X (gfx1250) — compile-verified
//
#include <hip/hip_runtime.h>
#include <hip/hip_bf16.h>

typedef float v2f __attribute__((ext_vector_type(2)));
typedef float v8f __attribute__((ext_vector_type(8)));

#define NROWS   65536          // 16*4096
#define DIM     64
#define KCODES  1024           // padded (real 1023)
#define KREAL   1023
#define EPSV    1e-8f

// workspace layout (float units)
#define WS_DITH 0                               // 1024*64
#define WS_BIAS (WS_DITH + KCODES * DIM)        // 1024
#define WS_BBUF (WS_BIAS + KCODES)              // 1024*64 (v2f-swizzled B)
#define WS_IDX  (WS_BBUF + KCODES * DIM)        // 65536 ints
#define WS_ACC  (WS_IDX + NROWS)                // 1 float

#define ZQ_ELEMS (NROWS * DIM)
#define LOSS_OFF ZQ_ELEMS
#define IDX_OFF  (ZQ_ELEMS + 1)

__global__ void zero_acc_kernel(float* ws) {
    ws[WS_ACC] = 0.0f;
}

// One wave per code j: dithered row + bias = -0.5*||dithered_j||^2 (pad j=1023).
__global__ void prep_dith_kernel(const float* __restrict__ lam_pairs,
                                 const float* __restrict__ cb,
                                 float* __restrict__ ws) {
    int j = blockIdx.x;               // 0..1023
    int lane = threadIdx.x;           // 0..31
    float lam = (j < KREAL) ? lam_pairs[j] : 0.0f;
    float acc = 0.0f;
    float* dith = ws + WS_DITH;
    #pragma unroll
    for (int u = 0; u < 2; ++u) {
        int d = lane + 32 * u;
        float v = 0.0f;
        if (j < KREAL) {
            float ci  = cb[j * DIM + d];
            float cp  = cb[(j + 1) * DIM + d];
            v = (1.0f - lam) * ci + lam * cp;
        }
        dith[j * DIM + d] = v;
        acc += v * v;
    }
    #pragma unroll
    for (int m = 16; m >= 1; m >>= 1) acc += __shfl_xor(acc, m, 32);
    if (lane == 0) {
        ws[WS_BIAS + j] = (j < KREAL) ? (-0.5f * acc) : -1.0e30f;
    }
}

// Swizzle dithered into per-lane WMMA B layout:
// B2[(tile*16+kk)*32 + lane] = { dith[n][kb], dith[n][kb+1] },
// n = tile*16 + (lane&15), kb = 4*kk + 2*(lane>>4).
__global__ void prep_bbuf_kernel(float* __restrict__ ws) {
    int tid = blockIdx.x * blockDim.x + threadIdx.x;   // 0..32767
    int tile = tid >> 9;            // /512
    int rem  = tid & 511;
    int kk   = rem >> 5;
    int lane = rem & 31;
    int n  = tile * 16 + (lane & 15);
    int kb = 4 * kk + 2 * (lane >> 4);
    const float* dith = ws + WS_DITH;
    v2f b;
    b.x = dith[n * DIM + kb];
    b.y = dith[n * DIM + kb + 1];
    ((v2f*)(ws + WS_BBUF))[tid] = b;
}

// Main scoring kernel: one wave handles TWO 16-row blocks (32 rows) across all
// 1024 codes; each B tile load feeds 32 WMMAs (2x register blocking).
// score = x . dithered_j - 0.5*||dithered_j||^2 ; argmax == argmin of d2.
__global__ void __launch_bounds__(256)
score_argmax_kernel(const float* __restrict__ x,
                    float* __restrict__ ws,
                    float* __restrict__ out_idx_f) {
    int wave = threadIdx.x >> 5;
    int lane = threadIdx.x & 31;
    int rb0  = (blockIdx.x * 8 + wave) * 2;    // first of two row blocks
    int row0 = rb0 * 16;
    int m0    = row0 + (lane & 15);
    int m1    = m0 + 16;
    int khalf = 2 * (lane >> 4);

    // A tiles: 2 x (16x64 fp32) held in 2x16 v2f regs (ISA 32-bit A layout:
    // M = lane&15, K = 2*(lane>>4) + vgpr within each K=4 chunk).
    v2f a0[16], a1[16];
    #pragma unroll
    for (int kk = 0; kk < 16; ++kk) {
        a0[kk] = *(const v2f*)&x[(size_t)m0 * DIM + 4 * kk + khalf];
        a1[kk] = *(const v2f*)&x[(size_t)m1 * DIM + 4 * kk + khalf];
    }

    const v2f*   B2   = (const v2f*)(ws + WS_BBUF);
    const float* bias = ws + WS_BIAS;

    float bestv0[8], bestv1[8];
    int   besti0[8], besti1[8];
    #pragma unroll
    for (int r = 0; r < 8; ++r) {
        bestv0[r] = -3.0e38f; besti0[r] = 0;
        bestv1[r] = -3.0e38f; besti1[r] = 0;
    }

    for (int tile = 0; tile < 64; ++tile) {
        v8f c0 = {}, c1 = {};
        const v2f* bt = B2 + (size_t)tile * 512 + lane;
        #pragma unroll
        for (int kk = 0; kk < 16; ++kk) {
            v2f b = bt[kk * 32];
            c0 = __builtin_amdgcn_wmma_f32_16x16x4_f32(
                     false, a0[kk], false, b, (short)0, c0, false, false);
            c1 = __builtin_amdgcn_wmma_f32_16x16x4_f32(
                     false, a1[kk], false, b, (short)0, c1, false, false);
        }
        int   n  = tile * 16 + (lane & 15);
        float bn = bias[n];
        #pragma unroll
        for (int r = 0; r < 8; ++r) {
            float s0 = c0[r] + bn;
            if (s0 > bestv0[r]) { bestv0[r] = s0; besti0[r] = n; }
            float s1 = c1[r] + bn;
            if (s1 > bestv1[r]) { bestv1[r] = s1; besti1[r] = n; }
        }
    }

    // Reduce over the 16 lanes holding the same row (xor masks stay in-half).
    #pragma unroll
    for (int r = 0; r < 8; ++r) {
        #pragma unroll
        for (int msk = 1; msk <= 8; msk <<= 1) {
            float ov0 = __shfl_xor(bestv0[r], msk, 32);
            int   oi0 = __shfl_xor(besti0[r], msk, 32);
            if (ov0 > bestv0[r] || (ov0 == bestv0[r] && oi0 < besti0[r])) {
                bestv0[r] = ov0; besti0[r] = oi0;
            }
            float ov1 = __shfl_xor(bestv1[r], msk, 32);
            int   oi1 = __shfl_xor(besti1[r], msk, 32);
            if (ov1 > bestv1[r] || (ov1 == bestv1[r] && oi1 < besti1[r])) {
                bestv1[r] = ov1; besti1[r] = oi1;
            }
        }
    }
    if ((lane & 15) == 0) {
        // lanes 0 / 16 hold rows r / 8+r of each C tile (C/D layout).
        int base0 = row0 + (lane >> 4) * 8;
        int base1 = base0 + 16;
        int* idxbuf = (int*)(ws + WS_IDX);
        #pragma unroll
        for (int r = 0; r < 8; ++r) {
            idxbuf[base0 + r]    = besti0[r];
            out_idx_f[base0 + r] = (float)besti0[r];
            idxbuf[base1 + r]    = besti1[r];
            out_idx_f[base1 + r] = (float)besti1[r];
        }
    }
}

// z_q + loss: one wave per row (2 dims/lane).
__global__ void __launch_bounds__(256)
zq_loss_kernel(const float* __restrict__ x,
               const float* __restrict__ lam_pairs,
               const float* __restrict__ cb,
               float* __restrict__ ws,
               float* __restrict__ zq_out) {
    __shared__ float bsum;
    if (threadIdx.x == 0) bsum = 0.0f;
    __syncthreads();

    int wave = threadIdx.x >> 5;
    int lane = threadIdx.x & 31;
    int row  = blockIdx.x * 8 + wave;

    const int* idxbuf = (const int*)(ws + WS_IDX);
    int idx   = idxbuf[row];
    float lam = lam_pairs[idx];

    float xv[2], di[2], dp[2];
    float s_i = 0.0f, s_p = 0.0f, s_e = 0.0f;
    #pragma unroll
    for (int u = 0; u < 2; ++u) {
        int d = lane + 32 * u;
        float xx = x[(size_t)row * DIM + d];
        float ci = cb[idx * DIM + d];
        float cp = cb[(idx + 1) * DIM + d];
        float t  = (1.0f - lam) * ci + lam * cp;   // dithered[idx]
        float e  = t - xx;
        xv[u] = xx;
        di[u] = ci - xx;
        dp[u] = cp - xx;
        s_i += di[u] * di[u];
        s_p += dp[u] * dp[u];
        s_e += e * e;
    }
    #pragma unroll
    for (int m = 16; m >= 1; m >>= 1) {
        s_i += __shfl_xor(s_i, m, 32);
        s_p += __shfl_xor(s_p, m, 32);
        s_e += __shfl_xor(s_e, m, 32);
    }
    float mag_i = sqrtf(s_i);
    float mag_p = sqrtf(s_p);
    float w_i = (1.0f - lam) * mag_i / (mag_i + EPSV);
    float w_p = lam * mag_p / (mag_p + EPSV);
    #pragma unroll
    for (int u = 0; u < 2; ++u) {
        int d = lane + 32 * u;
        zq_out[(size_t)row * DIM + d] = xv[u] + w_i * di[u] + w_p * dp[u];
    }
    if (lane == 0) atomicAdd(&bsum, s_e);
    __syncthreads();
    if (threadIdx.x == 0) atomicAdd(ws + WS_ACC, bsum);
}

__global__ void finalize_loss_kernel(const float* __restrict__ ws,
                                     float* __restrict__ loss_out) {
    // loss = codebook_loss + 0.25*commitment_loss = 1.25 * mean((t-x)^2)
    loss_out[0] = ws[WS_ACC] * 1.25f / (float)(NROWS * DIM);
}

extern "C" void kernel_launch(void* const* d_in, const int* in_sizes, int n_in,
                              void* d_out, int out_size, void* d_ws, size_t ws_size,
                              hipStream_t stream) {
    const float* z   = (const float*)d_in[0];   // (16,4096,64)
    const float* lam = (const float*)d_in[1];   // (1023,1)
    const float* cb  = (const float*)d_in[2];   // (1024,64)
    float* out = (float*)d_out;
    float* ws  = (float*)d_ws;

    zero_acc_kernel<<<1, 1, 0, stream>>>(ws);
    prep_dith_kernel<<<KCODES, 32, 0, stream>>>(lam, cb, ws);
    prep_bbuf_kernel<<<128, 256, 0, stream>>>(ws);               // 32768 threads
    score_argmax_kernel<<<NROWS / 256, 256, 0, stream>>>(z, ws, out + IDX_OFF);
    zq_loss_kernel<<<NROWS / 8, 256, 0, stream>>>(z, lam, cb, ws, out);
    finalize_loss_kernel<<<1, 1, 0, stream>>>(ws, out + LOSS_OFF);
}